// unitary_est_80650895884930
// MI455X (gfx1250) — compile-verified
//
#include <hip/hip_runtime.h>
#include <math.h>

typedef __attribute__((ext_vector_type(2))) float v2f;
typedef __attribute__((ext_vector_type(8))) float v8f;

// D = A(16x4,f32) * B(4x16,f32) + C(16x16,f32)
#define WMMA4(a, b, c) \
  __builtin_amdgcn_wmma_f32_16x16x4_f32(false, (a), false, (b), (short)0, (c), false, false)

__device__ __forceinline__ void wave_lds_fence() {
  // LDS ops from one wave complete in order (ISA 7.1);
  // this only pins the compiler's instruction ordering.
  __builtin_amdgcn_wave_barrier();
}

// Complex 16x16 matmul T = A*B. A,B in LDS, interleaved re/im: elem(i,j) at [2*(16*i+j)](re), +1(im).
// Result in C-layout registers (v8f real, v8f imag).
__device__ __forceinline__ void cmatmul16(const float* A, const float* B, int lane,
                                          v8f& cr, v8f& ci) {
  const int mn   = lane & 15;          // A row (m) / B col (n) for this lane
  const int koff = (lane >> 4) << 1;   // lanes 16-31 hold K+2 (ISA 7.12.2, 32-bit A 16x4)
  v8f zr = {0, 0, 0, 0, 0, 0, 0, 0};
  cr = zr; ci = zr;
#pragma unroll
  for (int kk = 0; kk < 4; ++kk) {
    const int kb = (kk << 2) + koff;
    v2f ar, ai, br, bi, nai;
    const int a0 = 2 * ((mn << 4) + kb);       // A[m][kb], A[m][kb+1] adjacent complex
    ar.x = A[a0];     ai.x = A[a0 + 1];
    ar.y = A[a0 + 2]; ai.y = A[a0 + 3];
    const int b0 = 2 * ((kb << 4) + mn);       // B[kb][n]
    const int b1 = 2 * (((kb + 1) << 4) + mn); // B[kb+1][n]
    br.x = B[b0]; bi.x = B[b0 + 1];
    br.y = B[b1]; bi.y = B[b1 + 1];
    nai.x = -ai.x; nai.y = -ai.y;              // f32 WMMA has no A/B neg modifier
    cr = WMMA4(ar,  br, cr);                   // +Ar*Br
    cr = WMMA4(nai, bi, cr);                   // -Ai*Bi
    ci = WMMA4(ar,  bi, ci);                   // +Ar*Bi
    ci = WMMA4(ai,  br, ci);                   // +Ai*Br
  }
}

// Fused: H = x * sigma (WMMA GEMM, H kept in LDS), then expm(-H) per batch
// via scaling-and-squaring + order-8 Taylor-Horner, all matmuls on v_wmma_f32_16x16x4_f32.
__global__ __launch_bounds__(256) void su2n_expm_fused(const float* __restrict__ x,
                                                       const float* __restrict__ sig,
                                                       float* __restrict__ out) {
  __shared__ float Hs[16 * 512];   // 32 KB: H tile, later overwritten in place with M = -H*2^-s
  __shared__ float Pss[8 * 512];   // 16 KB: stage1 x-tile [16][256]; stage2 per-wave P buffers

  const int tid  = threadIdx.x;
  const int lane = tid & 31;
  const int w    = tid >> 5;
  const int b0   = blockIdx.x << 4;
  const int mn   = lane & 15;
  const int koff = (lane >> 4) << 1;
  const int mh   = (lane >> 4) << 3;   // C-layout: lanes 16-31 hold rows M+8

  // ---------------- Stage 1a: x tile -> LDS (zero-pad K: 255 -> 256) ----------------
  float* xs = Pss;  // [16][256] = 4096 floats, aliases stage-2 P buffers
  for (int idx = tid; idx < 16 * 256; idx += 256) {
    const int m = idx >> 8, k = idx & 255;
    xs[idx] = (k < 255) ? x[(b0 + m) * 255 + k] : 0.0f;
  }
  __syncthreads();

  // ---------------- Stage 1b: H(16x512) = x_tile(16x256) * sigma(256x512) ----------------
  // sigma's natural complex-interleaved flat layout IS the B matrix: sig[k*512 + c].
#pragma unroll
  for (int tt = 0; tt < 4; ++tt) {
    const int c = (((w << 2) + tt) << 4) + mn;
    v8f acc = {0, 0, 0, 0, 0, 0, 0, 0};
    // Main loop: k-chunks 0..62 are fully in-bounds (kb+1 <= 251) -> no guards, no branches.
    for (int kk = 0; kk < 63; ++kk) {
      const int kb = (kk << 2) + koff;
      v2f a, b;
      a.x = xs[(mn << 8) + kb];
      a.y = xs[(mn << 8) + kb + 1];
      b.x = sig[kb * 512 + c];
      b.y = sig[(kb + 1) * 512 + c];
      acc = WMMA4(a, b, acc);
    }
    // Peeled last chunk (kb = 252 or 254): clamp address + branch-free select for the k=255 pad.
    {
      const int kb = 252 + koff;
      v2f a, b;
      a.x = xs[(mn << 8) + kb];
      a.y = xs[(mn << 8) + kb + 1];            // xs column 255 is zero-padded
      b.x = sig[kb * 512 + c];
      const int k1 = (kb + 1 < 255) ? (kb + 1) : 254;   // safe address for all lanes
      const float by = sig[k1 * 512 + c];
      b.y = (kb + 1 < 255) ? by : 0.0f;        // v_cndmask, no EXEC change
      acc = WMMA4(a, b, acc);
    }
#pragma unroll
    for (int r = 0; r < 8; ++r)
      Hs[(r + mh) * 512 + c] = acc[r];
  }
  __syncthreads();

  // ---------------- Stage 2: expm(-H) per batch; each wave handles 2 batches ----------------
  float* Ps = Pss + (w << 9);  // this wave's 512-float P buffer
  for (int rep = 0; rep < 2; ++rep) {
    const int bl = w + (rep << 3);
    float* Ms = Hs + (bl << 9);  // H[bl], scaled in place to M

    // Frobenius norm^2 (wave-wide butterfly reduce -> uniform in all lanes)
    float ss = 0.0f;
#pragma unroll
    for (int t = 0; t < 16; ++t) { const float v = Ms[lane + (t << 5)]; ss += v * v; }
#pragma unroll
    for (int d = 16; d >= 1; d >>= 1) ss += __shfl_xor(ss, d);
    const float fro = sqrtf(ss);
    int s = 0;
    if (fro > 0.25f) {
      s = (int)ceilf(log2f(fro * 4.0f));      // ||M|| <= 0.25 after scaling
      s = s < 0 ? 0 : (s > 30 ? 30 : s);
    }
    const float msc = -exp2f(-(float)s);      // M = -H * 2^-s

    // M in place; P = I + M/8 (order-8 Taylor seed)
#pragma unroll
    for (int q = 0; q < 16; ++q) {
      const int cidx = lane + (q << 5);
      const float mval = msc * Ms[cidx];
      Ms[cidx] = mval;
      const int pos = cidx >> 1;
      const bool diag_re = ((cidx & 1) == 0) && ((pos & 15) == (pos >> 4));
      Ps[cidx] = mval * 0.125f + (diag_re ? 1.0f : 0.0f);
    }
    wave_lds_fence();

    // Horner: P = I + (M*P)/j , j = 7..1
    for (int j = 7; j >= 1; --j) {
      v8f cr, ci;
      cmatmul16(Ms, Ps, lane, cr, ci);
      const float rj = 1.0f / (float)j;
      wave_lds_fence();
#pragma unroll
      for (int r = 0; r < 8; ++r) {
        const int m   = r + mh;
        const int idx = 2 * ((m << 4) + mn);
        Ps[idx]     = cr[r] * rj + ((m == mn) ? 1.0f : 0.0f);
        Ps[idx + 1] = ci[r] * rj;
      }
      wave_lds_fence();
    }

    // Squaring: P <- P*P, s times (s is wave-uniform -> EXEC stays all-1s)
    for (int t = 0; t < s; ++t) {
      v8f cr, ci;
      cmatmul16(Ps, Ps, lane, cr, ci);
      wave_lds_fence();
#pragma unroll
      for (int r = 0; r < 8; ++r) {
        const int m   = r + mh;
        const int idx = 2 * ((m << 4) + mn);
        Ps[idx]     = cr[r];
        Ps[idx + 1] = ci[r];
      }
      wave_lds_fence();
    }

    // Output: complex64 interleaved, coalesced 8-byte stores
    const int b = b0 + bl;
    float2* op = (float2*)(out + (size_t)b * 512);
#pragma unroll
    for (int q = 0; q < 8; ++q) {
      const int e = lane + (q << 5);
      float2 v;
      v.x = Ps[2 * e];
      v.y = Ps[2 * e + 1];
      op[e] = v;
    }
  }
}

extern "C" void kernel_launch(void* const* d_in, const int* in_sizes, int n_in,
                              void* d_out, int out_size, void* d_ws, size_t ws_size,
                              hipStream_t stream) {
  const float* x   = (const float*)d_in[0];   // [B,255] f32
  const float* sig = (const float*)d_in[1];   // [255,16,16] complex64 -> [255][512] f32
  float* out = (float*)d_out;                 // [B,16,16] complex64 -> [B][512] f32
  const int K = 255;
  const int B = in_sizes[0] / K;              // 16384
  dim3 grid(B / 16), block(256);
  hipLaunchKernelGGL(su2n_expm_fused, grid, block, 0, stream, x, sig, out);
}